// MRPCEN_29789893165584
// MI455X (gfx1250) — compile-verified
//
#include <hip/hip_runtime.h>
#include <cstdint>

// MR-PCEN on MI455X (gfx1250).
//   x: [B=8, F=128, N=8192] f32 ; out: [B, T=4, F, N] f32
// One block per (b,f) row; 4 waves, wave w owns rate T_VALUES[w].
// Row staged once into LDS via CDNA5 async global->LDS copies (ASYNCcnt).
// Per 256-timestep tile the zero-carry local scans of 16 sub-chunks (16 steps
// each) are computed as a 16x16x16 triangular matmul on the matrix pipe:
// 4 chained V_WMMA_F32_16X16X4_F32. Carries composed with a 16-step scalar
// chain (decay a^16) via v_readlane, then fused PCEN pointwise using raw
// v_log_f32/v_exp_f32 (log2/exp2 — the ln2 factors cancel in the pow chain).

#define PCEN_EPS 1e-5f
#define PCEN_N   8192

typedef __attribute__((ext_vector_type(2))) float v2f;
typedef __attribute__((ext_vector_type(8))) float v8f;

__device__ __forceinline__ float readlane_f32(float v, int l) {
    return __int_as_float(__builtin_amdgcn_readlane(__float_as_int(v), l));
}

// (base)^(e) for base > 0, via raw hardware log2/exp2 (no ocml guards).
__device__ __forceinline__ float pow_pos(float base, float e) {
    return __builtin_amdgcn_exp2f(e * __builtin_amdgcn_logf(base));
}

__launch_bounds__(128)
__global__ void mrpcen_kernel(const float* __restrict__ x,
                              const float* __restrict__ log_alpha,
                              const float* __restrict__ log_delta,
                              const float* __restrict__ log_r,
                              float* __restrict__ out,
                              int Fbands, int Tn)
{
    __shared__ __align__(16) float lds_x[PCEN_N];

    const int tid  = threadIdx.x;      // 0..127
    const int lane = tid & 31;
    const int wave = tid >> 5;         // 0..3 -> T index
    const int row  = blockIdx.x;       // b*F + f
    const int f    = row % Fbands;

    const uint64_t rowbase = (uint64_t)(uintptr_t)(x + (size_t)row * PCEN_N);

    // ---- Phase 1: async-stage the whole 32KB row into LDS (ASYNCcnt path) ----
    {
        const uint32_t lbase = (uint32_t)(uintptr_t)(&lds_x[0]);
        #pragma unroll
        for (int k = 0; k < 16; ++k) {
            uint32_t goff  = (uint32_t)(tid * 16 + k * 2048);  // byte offset in row
            uint32_t laddr = lbase + goff;                     // same offset in LDS
            asm volatile("global_load_async_to_lds_b128 %0, %1, %2 offset:0"
                         :: "v"(laddr), "v"(goff), "s"(rowbase)
                         : "memory");
        }
        asm volatile("s_wait_asynccnt 0" ::: "memory");
    }
    __syncthreads();

    // ---- Per-wave smoothing constants ----
    const float tv = (wave == 0) ? 2.0f : (wave == 1) ? 8.0f
                   : (wave == 2) ? 32.0f : 128.0f;
    const float s  = (sqrtf(1.0f + 4.0f * tv * tv) - 1.0f) / (2.0f * tv * tv);
    const float a  = 1.0f - s;
    const float a2 = a * a, a4 = a2 * a2, a8 = a4 * a4;
    const float A16 = a8 * a8;                 // per-sub-chunk decay (16 steps)

    const int n    = lane & 15;                // N / position within sub-chunk
    const int half = lane >> 4;                // 0: lanes 0-15, 1: lanes 16-31
    const int j0   = half ? 2 : 0;             // A/B K-index pair for this lane

    // B_k (4x16) decay matrices, per-wave constants, WMMA B layout:
    //   vgpr0: row K=j0, col N=n ; vgpr1: row K=j0+1.
    //   B_k[j, i] = s * a^(i - 4k - j)  for (4k+j) <= i, else 0.
    v2f Bk[4];
    #pragma unroll
    for (int k = 0; k < 4; ++k) {
        const int e0 = n - 4 * k - j0;
        const int e1 = e0 - 1;
        Bk[k].x = (e0 >= 0) ? s * pow_pos(a, (float)e0) : 0.0f;
        Bk[k].y = (e1 >= 0) ? s * pow_pos(a, (float)e1) : 0.0f;
    }
    const float ai1 = pow_pos(a, (float)(n + 1));  // a^(n+1) carry fix-up

    // Per-band parameters (stored in log space)
    const float la     = log_alpha[f];
    const float ldl    = log_delta[f];
    const float lrr    = log_r[f];
    const float alpha  = __expf(la);
    const float delta  = __expf(ldl);
    const float rr     = __expf(lrr);
    const float deltar = __expf(rr * ldl);     // delta^r = exp(r * log_delta)
    const float nalpha = -alpha;

    float C = lds_x[0];                        // m[-1] = x[0]

    float* orow = out + ((((size_t)(row / Fbands)) * Tn + wave) * Fbands + f)
                        * (size_t)PCEN_N;

    const int hoff = half << 7;                // 128*half: row offset of D rows

    // ---- Main loop: 32 tiles of 256 timesteps ----
    for (int tile = 0; tile < 32; ++tile) {
        const int tb = tile << 8;

        // Local (zero-carry) scans of 16 sub-chunks x 16 positions:
        //   D[c, i] = sum_{j<=i} x[tb + 16c + j] * s * a^(i-j)
        // A_k layout (16x4 f32): lane holds M = n, K = {j0, j0+1}.
        v8f D = {};
        #pragma unroll
        for (int k = 0; k < 4; ++k) {
            const v2f Ak = *reinterpret_cast<const v2f*>(
                &lds_x[tb + n * 16 + 4 * k + j0]);
            D = __builtin_amdgcn_wmma_f32_16x16x4_f32(
                    false, Ak, false, Bk[k], (short)0, D, false, false);
        }

        // Compose sub-chunk carries: K_0 = C ; K_{c+1} = D[c,15] + K_c * a^16.
        // D[c,15] sits in lane 15 (c=0..7, vgpr c) / lane 31 (c=8..15, vgpr c-8).
        float KLo[8], KHi[8];
        float Kc = C;
        #pragma unroll
        for (int c = 0; c < 16; ++c) {
            if (c < 8) KLo[c] = Kc; else KHi[c - 8] = Kc;
            const float ce = readlane_f32(D[c & 7], (c < 8) ? 15 : 31);
            Kc = fmaf(Kc, A16, ce);
        }
        C = Kc;                                // carry into next 256-step tile

        // Fix-up + fused PCEN pointwise + store.
        // D vgpr r, this lane: element (c = r + 8*half, i = n),
        //   global index = tb + 16c + n = tb + 16r + 128*half + n.
        #pragma unroll
        for (int r = 0; r < 8; ++r) {
            const float Kr = half ? KHi[r] : KLo[r];
            const float m  = fmaf(Kr, ai1, D[r]);
            const int   gi = tb + (r << 4) + hoff + n;
            const float xv = lds_x[gi];
            // (eps+m)^(-alpha): m >= 0 so the log2 argument is positive.
            const float sm = pow_pos(m + PCEN_EPS, nalpha);
            // (x*sm + delta)^r - delta^r: argument >= delta > 0.
            orow[gi] = pow_pos(fmaf(xv, sm, delta), rr) - deltar;
        }
    }
}

extern "C" void kernel_launch(void* const* d_in, const int* in_sizes, int n_in,
                              void* d_out, int out_size, void* d_ws, size_t ws_size,
                              hipStream_t stream) {
    const float* x         = (const float*)d_in[0];
    const float* log_alpha = (const float*)d_in[1];
    const float* log_delta = (const float*)d_in[2];
    const float* log_r     = (const float*)d_in[3];
    float* out = (float*)d_out;

    const int F    = in_sizes[1];             // 128 bands
    const int rows = in_sizes[0] / PCEN_N;    // B*F = 1024
    const int T    = 4;

    mrpcen_kernel<<<rows, 128, 0, stream>>>(x, log_alpha, log_delta, log_r,
                                            out, F, T);
}